// VaSCL_NBiDir_10892037063160
// MI455X (gfx1250) — compile-verified
//
#include <hip/hip_runtime.h>

// ---------------------------------------------------------------------------
// VaSCL hard-negative contrastive loss, CDNA5 (gfx1250, wave32).
// Never forms the [2B,2B] similarity matrix: per anchor r, a 2x128
// gather-GEMM (q1,q2 vs 128 gathered hard-negative rows) computed with
// V_WMMA_F32_16X16X4_F32 (exact fp32 — exp(x/0.05) amplifies dot error 20x).
// A-fragment comes from a zero-padded 3-row LDS block so every lane issues
// one unconditional ds_load_b64 per K-step (no EXEC-masked divergent loads).
// ---------------------------------------------------------------------------

typedef __attribute__((ext_vector_type(2))) float v2f;
typedef __attribute__((ext_vector_type(8))) float v8f;

#define B_ROWS 4096
#define D_DIM  256
#define K_HARD 64
#define INV_T  20.0f   // 1 / 0.05

__global__ __launch_bounds__(128)
void vascl_hardneg_kernel(const float* __restrict__ f1,
                          const float* __restrict__ f2,
                          const int*   __restrict__ hidx,
                          float*       __restrict__ per_row_loss)
{
    const int r    = blockIdx.x;
    const int tid  = threadIdx.x;
    const int lane = tid & 31;
    const int wave = tid >> 5;

    // Row 0 = q1, row 1 = q2, row 2 = zeros (read by lanes with M >= 2).
    __shared__ float sQ3[3][D_DIM];
    __shared__ int   sAdj[K_HARD];
    __shared__ float sNg[2][4];
    __shared__ float sPos;

    for (int d = tid; d < D_DIM; d += 128) {
        sQ3[0][d] = f1[(size_t)r * D_DIM + d];
        sQ3[1][d] = f2[(size_t)r * D_DIM + d];
        sQ3[2][d] = 0.0f;
    }
    if (tid < K_HARD) {
        int h = hidx[(size_t)r * K_HARD + tid];
        sAdj[tid] = h + (h >= r ? 1 : 0);   // skip removed diagonal column
    }
    __syncthreads();

    // pos = exp(<q1,q2>/T) on wave 0 (VALU + wave32 xor reduction).
    if (wave == 0) {
        float p = 0.f;
        for (int d = lane; d < D_DIM; d += 32)
            p += sQ3[0][d] * sQ3[1][d];
        for (int m = 16; m; m >>= 1)
            p += __shfl_xor(p, m, 32);
        if (lane == 0) sPos = expf(p * INV_T);
    }

    // ---- WMMA gather-GEMM ----
    // A (16x4 fp32): lanes 0-15 -> M=lane, K=k0,k0+1; lanes 16-31 -> M=lane-16,
    // K=k0+2,k0+3.  Rows M=0 (q1), M=1 (q2), M>=2 zero (LDS row 2).
    // B (4x16 fp32): mirrored -> lane half selects K pair, lane&15 = N column.
    const int half = lane >> 4;
    const int n    = lane & 15;
    const int koff = half * 2;
    const float* qrow = sQ3[n < 2 ? n : 2];      // zero row for M >= 2

    // This wave's two 16-column tiles (128 gathered columns across 4 waves).
    const int col0 = wave * 32 + n;              // tile 0
    const int col1 = col0 + 16;                  // tile 1
    const float* rowp0 = (col0 < K_HARD) ? (f1 + (size_t)sAdj[col0] * D_DIM)
                                         : (f2 + (size_t)sAdj[col0 - K_HARD] * D_DIM);
    const float* rowp1 = (col1 < K_HARD) ? (f1 + (size_t)sAdj[col1] * D_DIM)
                                         : (f2 + (size_t)sAdj[col1 - K_HARD] * D_DIM);

    // 4 independent accumulator chains: 2 tiles x even/odd K ping-pong.
    v8f cA0 = {}, cA1 = {}, cB0 = {}, cB1 = {};

    #pragma unroll 8
    for (int kk = 0; kk < D_DIM / 4; kk += 2) {
        const int ka = kk * 4 + koff;
        const int kb = ka + 4;

        // One A fragment per K-step, shared by both tiles.
        v2f a0 = *(const v2f*)(qrow + ka);       // ds_load_b64, 8B aligned
        v2f a1 = *(const v2f*)(qrow + kb);
        v2f b0t0 = *(const v2f*)(rowp0 + ka);    // global_load_b64 (L2 hits)
        v2f b1t0 = *(const v2f*)(rowp0 + kb);
        v2f b0t1 = *(const v2f*)(rowp1 + ka);
        v2f b1t1 = *(const v2f*)(rowp1 + kb);

        cA0 = __builtin_amdgcn_wmma_f32_16x16x4_f32(false, a0, false, b0t0,
                                                    (short)0, cA0, false, false);
        cB0 = __builtin_amdgcn_wmma_f32_16x16x4_f32(false, a0, false, b0t1,
                                                    (short)0, cB0, false, false);
        cA1 = __builtin_amdgcn_wmma_f32_16x16x4_f32(false, a1, false, b1t0,
                                                    (short)0, cA1, false, false);
        cB1 = __builtin_amdgcn_wmma_f32_16x16x4_f32(false, a1, false, b1t1,
                                                    (short)0, cB1, false, false);
    }

    // D tile: VGPR v, lanes 0-15 -> M=v, N=lane.  Need M=0 (q1), M=1 (q2).
    float d1t0 = cA0[0] + cA1[0];
    float d2t0 = cA0[1] + cA1[1];
    float d1t1 = cB0[0] + cB1[0];
    float d2t1 = cB0[1] + cB1[1];

    // Ng sums over all columns -> add both tiles' exponentials pre-reduction.
    float e1 = (half == 0) ? (expf(d1t0 * INV_T) + expf(d1t1 * INV_T)) : 0.f;
    float e2 = (half == 0) ? (expf(d2t0 * INV_T) + expf(d2t1 * INV_T)) : 0.f;
    for (int m = 16; m; m >>= 1) {
        e1 += __shfl_xor(e1, m, 32);
        e2 += __shfl_xor(e2, m, 32);
    }

    if (lane == 0) { sNg[0][wave] = e1; sNg[1][wave] = e2; }
    __syncthreads();

    if (tid == 0) {
        float Ng1 = sNg[0][0] + sNg[0][1] + sNg[0][2] + sNg[0][3];
        float Ng2 = sNg[1][0] + sNg[1][1] + sNg[1][2] + sNg[1][3];
        float pos = sPos;
        per_row_loss[r] = -logf(pos / (Ng1 + pos)) - logf(pos / (Ng2 + pos));
    }
}

// Deterministic final mean over the 4096 per-anchor (row-pair) losses.
__global__ __launch_bounds__(256)
void vascl_reduce_kernel(const float* __restrict__ per_row_loss,
                         float* __restrict__ out)
{
    __shared__ float s[256];
    float acc = 0.f;
    for (int i = threadIdx.x; i < B_ROWS; i += 256)
        acc += per_row_loss[i];
    s[threadIdx.x] = acc;
    __syncthreads();
    for (int m = 128; m; m >>= 1) {
        if (threadIdx.x < m) s[threadIdx.x] += s[threadIdx.x + m];
        __syncthreads();
    }
    if (threadIdx.x == 0)
        out[0] = s[0] / (2.0f * (float)B_ROWS);
}

extern "C" void kernel_launch(void* const* d_in, const int* in_sizes, int n_in,
                              void* d_out, int out_size, void* d_ws, size_t ws_size,
                              hipStream_t stream)
{
    const float* f1   = (const float*)d_in[0];
    const float* f2   = (const float*)d_in[1];
    const int*   hidx = (const int*)d_in[2];
    float* per_row = (float*)d_ws;           // 4096 floats of scratch

    vascl_hardneg_kernel<<<B_ROWS, 128, 0, stream>>>(f1, f2, hidx, per_row);
    vascl_reduce_kernel<<<1, 256, 0, stream>>>(per_row, (float*)d_out);
}